// TopkLayer1d_83434034692100
// MI455X (gfx1250) — compile-verified
//
#include <hip/hip_runtime.h>

typedef float v2f __attribute__((ext_vector_type(2)));
typedef float v4f __attribute__((ext_vector_type(4)));
typedef float v8f __attribute__((ext_vector_type(8)));

#define NEURONS 128
#define WSIZE   64

// One zone per workgroup iteration. 256 threads = 8 wave32; wave w computes
// neurons [16w, 16w+16) of the zone via 16 chained v_wmma_f32_16x16x4_f32.
// B (the x-window) is replicated across the 16 WMMA columns, so every output
// column of the 16x16 tile equals the GEMV result for its row.
//
// W (535MB) is streamed exactly once -> non-temporal loads (TH=NT) so the
// 192MB L2 is reserved for x (64KB, heavily reused across overlapping
// windows) instead of being thrashed by dead weight data.
__global__ __launch_bounds__(256) void topk_wta_kernel(
    const float* __restrict__ x,
    const float* __restrict__ W,
    float* __restrict__ out,
    int zones) {
  __shared__ __attribute__((aligned(16))) float xwin[WSIZE];
  __shared__ float wavemax[8];

  const int tid  = threadIdx.x;
  const int wave = tid >> 5;
  const int lane = tid & 31;
  const int half = lane >> 4;   // K-pair select: 0 -> K{0,1}, 1 -> K{2,3}
  const int mrow = lane & 15;   // A-matrix row within the 16-row block

  for (int z = blockIdx.x; z < zones; z += gridDim.x) {
    // Stage the 64-float sliding window into LDS (coalesced; x lives in L2).
    if (tid < WSIZE) xwin[tid] = x[z + tid];

    // Prefetch next zone's 32KB weight slab (256 threads x 128B lines)
    // -> global_prefetch_b8; speculative, hides HBM latency across the loop.
    {
      int zn = z + (int)gridDim.x;
      if (zn < zones) {
        __builtin_prefetch(W + (size_t)zn * (NEURONS * WSIZE) + tid * 32, 0, 1);
      }
    }
    __syncthreads();

    const float* Arow =
        W + (size_t)z * (NEURONS * WSIZE)        // zone slab
        + (size_t)(wave * 16 + mrow) * WSIZE     // neuron row
        + half * 2;                              // K-pair within each chunk of 4

    v8f acc = {};
#pragma unroll
    for (int kb = 0; kb < 16; ++kb) {
      // A fragment: 2 consecutive f32 of this row, streamed non-temporally
      // (GLOBAL_LOAD_B64, TH=NT: read-once data, don't pollute L2).
      v2f a = __builtin_nontemporal_load((const v2f*)(Arow + kb * 4));
      // B fragment: matching window pair, same for all 16 lanes of a half
      // (DS_LOAD_B64 broadcast) -> replicates the window across all 16 cols.
      v2f b = *(const v2f*)(&xwin[kb * 4 + half * 2]);
      acc = __builtin_amdgcn_wmma_f32_16x16x4_f32(
          /*neg_a=*/false, a, /*neg_b=*/false, b,
          /*c_mod=*/(short)0, acc, /*reuse_a=*/false, /*reuse_b=*/false);
    }

    // C/D layout: vgpr r holds row (wave*16 + half*8 + r), replicated over
    // the 16 lanes of this half. Top-1 (K=1) winner-take-all:
    float lmax = acc[0];
#pragma unroll
    for (int r = 1; r < 8; ++r) lmax = fmaxf(lmax, acc[r]);
    lmax = fmaxf(lmax, __shfl_xor(lmax, 16, 32));   // merge the two halves
    if (lane == 0) wavemax[wave] = lmax;            // 16-row block max
    __syncthreads();

    float zmax = wavemax[0];
#pragma unroll
    for (int wv = 1; wv < 8; ++wv) zmax = fmaxf(zmax, wavemax[wv]);

    // Only lanes 0 and 16 write (values are replicated across 16 lanes):
    // two masked non-temporal float4 stores covering rows
    // [wave*16 + half*8, +8). Output is write-once -> TH=NT.
    if (mrow == 0) {
      float* dst = out + (size_t)z * NEURONS + wave * 16 + half * 8;
      v4f lo, hi;
      lo.x = (acc[0] >= zmax) ? acc[0] : 0.0f;
      lo.y = (acc[1] >= zmax) ? acc[1] : 0.0f;
      lo.z = (acc[2] >= zmax) ? acc[2] : 0.0f;
      lo.w = (acc[3] >= zmax) ? acc[3] : 0.0f;
      hi.x = (acc[4] >= zmax) ? acc[4] : 0.0f;
      hi.y = (acc[5] >= zmax) ? acc[5] : 0.0f;
      hi.z = (acc[6] >= zmax) ? acc[6] : 0.0f;
      hi.w = (acc[7] >= zmax) ? acc[7] : 0.0f;
      __builtin_nontemporal_store(lo, (v4f*)(dst));
      __builtin_nontemporal_store(hi, (v4f*)(dst + 4));
    }
  }
}

extern "C" void kernel_launch(void* const* d_in, const int* in_sizes, int n_in,
                              void* d_out, int out_size, void* d_ws, size_t ws_size,
                              hipStream_t stream) {
  (void)n_in; (void)out_size; (void)d_ws; (void)ws_size;
  const float* x = (const float*)d_in[0];
  const float* W = (const float*)d_in[1];
  float* out = (float*)d_out;

  const int zones = in_sizes[1] / (NEURONS * WSIZE);  // 16321
  int grid = zones < 4096 ? zones : 4096;             // grid-stride -> prefetch works
  topk_wta_kernel<<<grid, 256, 0, stream>>>(x, W, out, zones);
}